// IPLayer_74397423501698
// MI455X (gfx1250) — compile-verified
//
#include <hip/hip_runtime.h>
#include <hip/hip_bf16.h>
#include <stdint.h>

// ---------------------------------------------------------------------------
// segment_sum(inter[P,64], ind_2[:,0], n_atoms) on MI455X (gfx1250, wave32).
//
// Roofline: ~870 MB mandatory traffic @ 23.3 TB/s => ~37 us. Memory/atomic
// bound, so: build CSR (6.4M u32 atomics instead of 204.8M f32 atomics),
// then an atomic-free gather-reduce, one wave32 per atom.
//
// The gather uses the CDNA5 ASYNCcnt path: GLOBAL_LOAD_ASYNC_TO_LDS_B64
// stages one full 256B inter row per instruction into a per-wave LDS ring
// (no dest VGPRs), 8 rows in flight per wave, consumed via ds_load_b64.
// ---------------------------------------------------------------------------

#define ASYNC_DEPTH 8  // rows in flight per wave (power of two)

#if defined(__gfx1250__)
#define ASYNC_LOAD_ROW_B64(ldsoff, gaddr)                                     \
    asm volatile("global_load_async_to_lds_b64 %0, %1, off" ::"v"(ldsoff),    \
                 "v"(gaddr)                                                   \
                 : "memory")
#define WAIT_ASYNCCNT(n) asm volatile("s_wait_asynccnt " #n ::: "memory")
#define WAIT_DSCNT0() asm volatile("s_wait_dscnt 0x0" ::: "memory")
#endif

__global__ void k_zero_u32(uint32_t* __restrict__ p, int n) {
    int i = blockIdx.x * blockDim.x + threadIdx.x;
    if (i < n) p[i] = 0u;
}

__global__ void k_zero_f32(float* __restrict__ p, int n) {
    int i = blockIdx.x * blockDim.x + threadIdx.x;
    if (i < n) p[i] = 0.0f;
}

// Pass 1: histogram of first atom index per pair.
__global__ void k_hist(const int2* __restrict__ ind2, uint32_t* __restrict__ cnt,
                       int n_pairs) {
    int p = blockIdx.x * blockDim.x + threadIdx.x;
    if (p < n_pairs) {
        int a = ind2[p].x;
        atomicAdd(&cnt[a], 1u);
    }
}

// Pass 2: single-workgroup exclusive scan, thread-coarsened x16 (7 chunks for
// 100k counters, ~30 barriers total). cnt[] is rewritten in place to the
// exclusive offsets (becomes the scatter cursor); offs[] additionally gets
// offs[n] = total.
#define SCAN_COARSE 16
__global__ __launch_bounds__(1024) void k_scan(uint32_t* __restrict__ cnt,
                                               uint32_t* __restrict__ offs,
                                               int n) {
    __shared__ uint32_t wsum[32];
    __shared__ uint32_t sbase;
    const int tid  = threadIdx.x;
    const int lane = tid & 31;
    const int wid  = tid >> 5;
    if (tid == 0) sbase = 0u;
    __syncthreads();

    const int chunk = 1024 * SCAN_COARSE;
    for (int base = 0; base < n; base += chunk) {
        uint32_t loc[SCAN_COARSE];
        uint32_t sum = 0u;
        const int i0 = base + tid * SCAN_COARSE;
#pragma unroll
        for (int j = 0; j < SCAN_COARSE; ++j) {
            uint32_t v = (i0 + j < n) ? cnt[i0 + j] : 0u;
            loc[j] = v;
            sum += v;
        }
        uint32_t incl = sum;  // wave32 inclusive scan of per-thread sums
#pragma unroll
        for (int d = 1; d < 32; d <<= 1) {
            uint32_t t = __shfl_up(incl, d, 32);
            if (lane >= d) incl += t;
        }
        if (lane == 31) wsum[wid] = incl;
        __syncthreads();
        if (wid == 0) {
            uint32_t w = wsum[lane];
            uint32_t s = w;
#pragma unroll
            for (int d = 1; d < 32; d <<= 1) {
                uint32_t t = __shfl_up(s, d, 32);
                if (lane >= d) s += t;
            }
            wsum[lane] = s - w;  // exclusive wave base
        }
        __syncthreads();
        const uint32_t tb = sbase + wsum[wid] + (incl - sum);  // thread base
        uint32_t run = tb;
#pragma unroll
        for (int j = 0; j < SCAN_COARSE; ++j) {
            uint32_t e = run;
            run += loc[j];
            if (i0 + j < n) { offs[i0 + j] = e; cnt[i0 + j] = e; }
        }
        __syncthreads();                    // all sbase reads done
        if (tid == 1023) sbase = tb + sum;  // running total
        __syncthreads();
    }
    if (tid == 0) offs[n] = sbase;          // == n_pairs
}

// Pass 3: bucket pair ids by atom.
__global__ void k_scatter(const int2* __restrict__ ind2,
                          uint32_t* __restrict__ cursor,
                          uint32_t* __restrict__ pidx, int n_pairs) {
    int p = blockIdx.x * blockDim.x + threadIdx.x;
    if (p < n_pairs) {
        int a = ind2[p].x;
        uint32_t pos = atomicAdd(&cursor[a], 1u);
        pidx[pos] = (uint32_t)p;
    }
}

// Pass 4: one wave32 per atom, lane handles cols {2L, 2L+1}. Atomic-free,
// barrier-free. Rows staged global->LDS with the ASYNCcnt DMA path, 8 deep.
__global__ __launch_bounds__(256) void k_gather64(
    const float* __restrict__ inter, const uint32_t* __restrict__ offs,
    const uint32_t* __restrict__ pidx, float2* __restrict__ out2, int n_atoms) {
    const int lane = threadIdx.x & 31;
    const int wid  = threadIdx.x >> 5;
    const int atom = blockIdx.x * (blockDim.x >> 5) + wid;

#if defined(__gfx1250__)
    // Per-wave ring: ASYNC_DEPTH rows x 64 floats. 8 waves -> 16 KB LDS.
    __shared__ float2 ring[256 / 32][ASYNC_DEPTH][32];

    if (atom >= n_atoms) return;
    const uint32_t s = offs[atom];
    const uint32_t e = offs[atom + 1];
    const uint32_t deg = e - s;

    // Per-lane constant parts of the LDS / global addresses.
    const uint32_t lds0 = (uint32_t)(uintptr_t)&ring[wid][0][lane];
    const uint64_t gab  = (uint64_t)(uintptr_t)inter + ((uint64_t)lane << 3);

    // Head: fill the pipeline.
    const uint32_t head = (deg < ASYNC_DEPTH) ? deg : (uint32_t)ASYNC_DEPTH;
    for (uint32_t d = 0; d < head; ++d) {
        uint32_t p = pidx[s + d];
        ASYNC_LOAD_ROW_B64(lds0 + (d << 8), gab + ((uint64_t)p << 8));
    }

    float ax = 0.0f, ay = 0.0f;
    uint32_t k = 0;
    // Steady state: exactly ASYNC_DEPTH rows outstanding at loop entry.
    // Completions are in-order, so asynccnt<=DEPTH-1 => oldest row landed.
    for (; k + ASYNC_DEPTH < deg; ++k) {
        __builtin_prefetch(&pidx[s + k + 64], 0, 1);  // -> global_prefetch_b8
        WAIT_ASYNCCNT(7);
        float2 r = ring[wid][k & (ASYNC_DEPTH - 1)][lane];
        ax += r.x;
        ay += r.y;
        uint32_t p = pidx[s + k + ASYNC_DEPTH];
        WAIT_DSCNT0();  // slot fully read before the DMA rewrites it
        ASYNC_LOAD_ROW_B64(lds0 + ((k & (ASYNC_DEPTH - 1)) << 8),
                           gab + ((uint64_t)p << 8));
    }
    // Drain: wait for everything, then consume the tail from LDS.
    WAIT_ASYNCCNT(0);
    for (; k < deg; ++k) {
        float2 r = ring[wid][k & (ASYNC_DEPTH - 1)][lane];
        ax += r.x;
        ay += r.y;
    }
    out2[(size_t)atom * 32 + lane] = make_float2(ax, ay);
#else
    // Host-pass / non-gfx1250 fallback: plain VGPR gather.
    if (atom >= n_atoms) return;
    const uint32_t s = offs[atom];
    const uint32_t e = offs[atom + 1];
    const float2* inter2 = (const float2*)inter;
    float ax = 0.0f, ay = 0.0f;
    for (uint32_t k = s; k < e; ++k) {
        uint32_t p = pidx[k];
        float2 r = inter2[(size_t)p * 32 + lane];
        ax += r.x;
        ay += r.y;
    }
    out2[(size_t)atom * 32 + lane] = make_float2(ax, ay);
#endif
}

// Fallback (only if ws too small / unexpected shape): direct atomic scatter.
__global__ void k_atomic_scatter(const int* __restrict__ ind2flat,
                                 const float* __restrict__ inter,
                                 float* __restrict__ out,
                                 int n_pairs, int n_cols) {
    long long t = (long long)blockIdx.x * blockDim.x + threadIdx.x;
    long long total = (long long)n_pairs * n_cols;
    if (t < total) {
        int p = (int)(t / n_cols);
        int c = (int)(t - (long long)p * n_cols);
        int a = ind2flat[2 * p];
        atomicAdd(&out[(size_t)a * n_cols + c], inter[t]);
    }
}

extern "C" void kernel_launch(void* const* d_in, const int* in_sizes, int n_in,
                              void* d_out, int out_size, void* d_ws, size_t ws_size,
                              hipStream_t stream) {
    const int2*  ind2  = (const int2*)d_in[0];   // [P,2] int32
    const float* inter = (const float*)d_in[2];  // [P,64] f32   (d_in[1]=prop unused)
    float*       out   = (float*)d_out;          // [n_atoms,64] f32

    const int n_pairs = in_sizes[0] / 2;
    const int n_cols  = in_sizes[2] / n_pairs;   // 64
    const int n_atoms = out_size / n_cols;       // 100000

    // Workspace layout (256B-aligned regions): offs | cursor/counts | pairIdx
    const size_t offsB = (((size_t)(n_atoms + 1) * 4) + 255) & ~(size_t)255;
    const size_t curB  = (((size_t)n_atoms * 4) + 255) & ~(size_t)255;
    const size_t need  = offsB + curB + (size_t)n_pairs * 4;

    if (n_cols == 64 && ws_size >= need) {
        uint32_t* offs = (uint32_t*)d_ws;
        uint32_t* cnt  = (uint32_t*)((char*)d_ws + offsB);  // counts -> cursor
        uint32_t* pidx = (uint32_t*)((char*)d_ws + offsB + curB);

        k_zero_u32<<<(n_atoms + 255) / 256, 256, 0, stream>>>(cnt, n_atoms);
        k_hist   <<<(n_pairs + 255) / 256, 256, 0, stream>>>(ind2, cnt, n_pairs);
        k_scan   <<<1, 1024, 0, stream>>>(cnt, offs, n_atoms);
        k_scatter<<<(n_pairs + 255) / 256, 256, 0, stream>>>(ind2, cnt, pidx, n_pairs);
        // gather writes every output row -> no out zero-init needed
        k_gather64<<<(n_atoms + 7) / 8, 256, 0, stream>>>(
            inter, offs, pidx, (float2*)out, n_atoms);
    } else {
        k_zero_f32<<<(out_size + 255) / 256, 256, 0, stream>>>(out, out_size);
        long long total = (long long)n_pairs * n_cols;
        int blocks = (int)((total + 255) / 256);
        k_atomic_scatter<<<blocks, 256, 0, stream>>>(
            (const int*)d_in[0], inter, out, n_pairs, n_cols);
    }
}